// CurvStdDist_24790551233441
// MI455X (gfx1250) — compile-verified
//
#include <hip/hip_runtime.h>
#include <math.h>

// Problem constants (match reference setup_inputs / K_NEIGHBORS)
#define NPTS  4096
#define NB    4
#define KNN   16
#define KLIST 17            // k+1, includes self
#define ROWPAD 20           // scratch row stride in floats (16B-aligned rows)

typedef float v2f __attribute__((ext_vector_type(2)));
typedef float v8f __attribute__((ext_vector_type(8)));

// Compiler-only memory fence: orders LDS stores/loads in the IR without
// emitting barriers. Hardware DS ops from one wave execute in order, so
// same-wave cross-lane RAW through per-wave LDS scratch is safe.
#define LDS_FENCE() asm volatile("" ::: "memory")

// ---------------------------------------------------------------------------
// Kernel A: brute-force 17-NN via WMMA f32 16x16x4 + kappa.
//
// Trick: with A(m,0:3)=(x,y,z,1), B(0:3,j)=(x,y,z,-0.5*|p_j|^2) and the WMMA
// accumulator C(m,j) = -0.5*|p_m|^2, the matrix op directly yields
//     P(m,j) = dot(p_m,p_j) - 0.5|p_j|^2 - 0.5|p_m|^2 = -0.5 * d2(m,j)
// so the entire pairwise-distance computation is fused into v_wmma, and
// top-K runs on P with max-selection (monotone in d2), needing only a
// ds_load + compare per candidate.
//
// grid = (NPTS/256, NB, 2)  block = 256 (8 waves, wave32)
// Each wave owns 32 rows (two 16x16 WMMA tiles) and a private scratch tile,
// so no inter-wave synchronization is needed inside the 256-tile loop.
// Dynamic LDS: 4096 x float4 (x,y,z,-0.5*sq) = 64KB + 8*32*20 f32 = 20KB
// ---------------------------------------------------------------------------
__global__ void __launch_bounds__(256)
knn_kappa_kernel(const float* __restrict__ ori, const float* __restrict__ adv,
                 const float* __restrict__ nrm,
                 float* __restrict__ kappa_out, int* __restrict__ knn_out)
{
    const int z  = blockIdx.z;           // 0 = ori, 1 = adv
    const int bi = blockIdx.y;
    const float* pc = ((z == 0) ? ori : adv) + (size_t)bi * 3 * NPTS;

    extern __shared__ float4 smem[];
    float4* pts = smem;                          // [NPTS] (x,y,z,-0.5*sq)
    float*  sc  = (float*)(pts + NPTS);          // [8][32][ROWPAD] P scratch

    // Stage planar (3,n) -> AoS float4; w = -0.5*|p|^2 (never fed to A coords)
    for (int i = threadIdx.x; i < NPTS; i += 256) {
        float x = pc[i], y = pc[NPTS + i], zc = pc[2 * NPTS + i];
        pts[i] = make_float4(x, y, zc, -0.5f * (x * x + y * y + zc * zc));
    }
    __syncthreads();

    const int lane = threadIdx.x & 31;
    const int w    = threadIdx.x >> 5;
    const int h    = lane >> 4;          // coord-pair select: 0->(K0,K1) 1->(K2,K3)
    const int lm   = lane & 15;          // row/col within 16-tile
    const int r0   = blockIdx.x * 256 + w * 32;
    float* mySc    = sc + w * (32 * ROWPAD);

    // A fragments (16x4 f32 layout: lane%16 = M, lane/16 picks coord pair).
    // K3 of A is 1.0 so that B's K3 row contributes -0.5*sq_j.
    float4 pA0 = pts[r0 + lm];
    float4 pA1 = pts[r0 + 16 + lm];
    v2f a_lo, a_hi;
    a_lo.x = h ? pA0.z : pA0.x;  a_lo.y = h ? 1.0f : pA0.y;
    a_hi.x = h ? pA1.z : pA1.x;  a_hi.y = h ? 1.0f : pA1.y;

    // C fragments: lane holds col N=lane%16, VGPR r -> row M = r + 8*(lane/16).
    // C(m,*) = -0.5*sq_m (row-constant), loaded once, reused all 256 tiles.
    v8f c_lo, c_hi;
#pragma unroll
    for (int r = 0; r < 8; ++r) {
        c_lo[r] = pts[r0 + 8 * h + r].w;
        c_hi[r] = pts[r0 + 16 + 8 * h + r].w;
    }

    const int mrow = r0 + lane;          // row this lane owns for selection

    // Prefetch the normal components we only need after the 256-tile loop
    // (probe-confirmed: lowers to global_prefetch_b8 on gfx1250).
    const float* nb = nrm + (size_t)bi * 3 * NPTS;
    __builtin_prefetch(nb + mrow, 0, 3);
    __builtin_prefetch(nb + NPTS + mrow, 0, 3);
    __builtin_prefetch(nb + 2 * NPTS + mrow, 0, 3);

    // Register-resident top-17 list of LARGEST P (== smallest d2),
    // sorted descending; fully unrolled -> stays in VGPRs.
    float nd[KLIST]; int ni[KLIST];
#pragma unroll
    for (int t = 0; t < KLIST; ++t) { nd[t] = -3.4e38f; ni[t] = 0; }

    for (int jt = 0; jt < NPTS / 16; ++jt) {
        const int j0 = jt * 16;

        // B fragment (4x16): same lane rule; K3 = -0.5*sq_col.
        float4 pB = pts[j0 + lm];
        v2f bf;
        bf.x = h ? pB.z : pB.x;
        bf.y = h ? pB.w : pB.y;

        v8f plo = __builtin_amdgcn_wmma_f32_16x16x4_f32(
            false, a_lo, false, bf, (short)0, c_lo, false, false);
        v8f phi = __builtin_amdgcn_wmma_f32_16x16x4_f32(
            false, a_hi, false, bf, (short)0, c_hi, false, false);

        // Scatter tile to per-wave scratch: row-major [32][ROWPAD].
        const int rowbase = h * 8;
#pragma unroll
        for (int r = 0; r < 8; ++r) {
            mySc[(rowbase + r)      * ROWPAD + lm] = plo[r];
            mySc[(16 + rowbase + r) * ROWPAD + lm] = phi[r];
        }
        LDS_FENCE();   // order stores before this wave's row reads

        // Selection: lane scans its row as 4x ds_load_b128 + compares.
        const float4* rowp = (const float4*)(mySc + lane * ROWPAD);
#pragma unroll
        for (int q = 0; q < 4; ++q) {
            float4 sv = rowp[q];
            float se[4] = {sv.x, sv.y, sv.z, sv.w};
#pragma unroll
            for (int e = 0; e < 4; ++e) {
                float s = se[e];
                int   j = j0 + q * 4 + e;
                if (s > nd[KLIST - 1]) {
                    nd[KLIST - 1] = s; ni[KLIST - 1] = j;
#pragma unroll
                    for (int t = KLIST - 1; t > 0; --t) {
                        if (nd[t] > nd[t - 1]) {
                            float td = nd[t - 1]; nd[t - 1] = nd[t]; nd[t] = td;
                            int   ti = ni[t - 1]; ni[t - 1] = ni[t]; ni[t] = ti;
                        }
                    }
                }
            }
        }
        LDS_FENCE();   // order row reads before next iteration's stores
    }

    // kappa = mean_t |normalize(p_j - p_m) . n_m| over 16 neighbors (skip self)
    float4 pm = pts[mrow];
    float nx = nb[mrow], ny = nb[NPTS + mrow], nz = nb[2 * NPTS + mrow];
    float acc = 0.0f;
#pragma unroll
    for (int t = 1; t < KLIST; ++t) {
        float4 pj = pts[ni[t]];
        float vx = pj.x - pm.x, vy = pj.y - pm.y, vz = pj.z - pm.z;
        float len = sqrtf(vx * vx + vy * vy + vz * vz);
        float inv = 1.0f / fmaxf(len, 1e-12f);
        acc += fabsf(vx * nx + vy * ny + vz * nz) * inv;
    }

    const size_t base = ((size_t)z * NB + bi) * NPTS;
    kappa_out[base + mrow] = acc * (1.0f / KNN);
    int* ko = knn_out + (base + mrow) * KNN;
#pragma unroll
    for (int t = 1; t < KLIST; ++t) ko[t - 1] = ni[t];
}

// ---------------------------------------------------------------------------
// Kernel B: gather neighbor kappas, unbiased std (ddof=1), accumulate per-batch
// sum of squared differences (ori_std - adv_std + 1e-6)^2.
// ---------------------------------------------------------------------------
__global__ void __launch_bounds__(256)
std_diff_kernel(const float* __restrict__ kappa, const int* __restrict__ knn,
                float* __restrict__ bsum)
{
    int gid = blockIdx.x * blockDim.x + threadIdx.x;
    if (gid >= NB * NPTS) return;
    int bi = gid / NPTS;
    int i  = gid - bi * NPTS;

    float s[2];
#pragma unroll
    for (int zz = 0; zz < 2; ++zz) {
        const size_t base = ((size_t)zz * NB + bi) * NPTS;
        const float* kap  = kappa + base;
        const int*   id   = knn + (base + i) * KNN;
        float v[KNN]; float m = 0.0f;
#pragma unroll
        for (int t = 0; t < KNN; ++t) { v[t] = kap[id[t]]; m += v[t]; }
        m *= (1.0f / KNN);
        float var = 0.0f;
#pragma unroll
        for (int t = 0; t < KNN; ++t) { float d = v[t] - m; var += d * d; }
        s[zz] = sqrtf(var * (1.0f / (KNN - 1)));
    }
    float diff = s[0] - s[1] + 1e-6f;
    atomicAdd(&bsum[bi], diff * diff);
}

__global__ void zero_bsum_kernel(float* __restrict__ bsum)
{
    if (threadIdx.x < NB) bsum[threadIdx.x] = 0.0f;
}

__global__ void finalize_kernel(const float* __restrict__ bsum, float* __restrict__ out)
{
    if (threadIdx.x == 0) {
        float a = 0.0f;
        for (int i = 0; i < NB; ++i) a += sqrtf(bsum[i]);
        out[0] = a * (1.0f / NB);
    }
}

// ---------------------------------------------------------------------------
extern "C" void kernel_launch(void* const* d_in, const int* in_sizes, int n_in,
                              void* d_out, int out_size, void* d_ws, size_t ws_size,
                              hipStream_t stream)
{
    const float* ori = (const float*)d_in[0];   // (b,3,n) f32
    const float* adv = (const float*)d_in[1];   // (b,3,n) f32
    const float* nrm = (const float*)d_in[2];   // (b,3,n) f32
    float* out = (float*)d_out;                 // scalar f32

    // Workspace: kappa[2][NB][NPTS] f32, knn[2][NB][NPTS][KNN] i32, bsum[NB] f32
    float* kappa = (float*)d_ws;
    int*   knn   = (int*)(kappa + 2 * NB * NPTS);
    float* bsum  = (float*)(knn + (size_t)2 * NB * NPTS * KNN);

    zero_bsum_kernel<<<1, 32, 0, stream>>>(bsum);

    dim3 gridA(NPTS / 256, NB, 2);
    size_t shmem = (size_t)NPTS * sizeof(float4)
                 + (size_t)8 * 32 * ROWPAD * sizeof(float);
    knn_kappa_kernel<<<gridA, 256, shmem, stream>>>(ori, adv, nrm, kappa, knn);

    std_diff_kernel<<<(NB * NPTS + 255) / 256, 256, 0, stream>>>(kappa, knn, bsum);
    finalize_kernel<<<1, 32, 0, stream>>>(bsum, out);
}